// ForgetMult_16647293240039
// MI455X (gfx1250) — compile-verified
//
#include <hip/hip_runtime.h>
#include <cstdint>
#include <cstddef>

// ForgetMult recurrence: h_t = f_t*x_t + (1-f_t)*h_{t-1}
// (SEQ, BATCH, HIDDEN) = (2048, 16, 1024), fp32.
// 3-pass chunked parallel scan, async global->LDS double-buffered streaming.

#define SEQ_LEN   2048
#define BATCH     16
#define HIDDEN    1024
#define NCH       (BATCH * HIDDEN)        // 16384 channels
#define CHUNKS    16
#define CHUNK_LEN (SEQ_LEN / CHUNKS)      // 128 timesteps per chunk
#define TPB       256
#define VEC       2                        // float2 per thread
#define CH_PER_BLK (TPB * VEC)             // 512 channels per block
#define T_TILE    4                        // timesteps staged per LDS tile
#define N_TILES   (CHUNK_LEN / T_TILE)     // 32 tiles per chunk

// One lane copies 8 bytes (its float2 column) from global into LDS via the
// CDNA5 async-DMA path. Tracked by ASYNCcnt, not LOADcnt; no VGPR data.
__device__ __forceinline__ void async_copy_b64(uint32_t lds_off, const void* gaddr) {
  asm volatile("global_load_async_to_lds_b64 %0, %1, off"
               :: "v"(lds_off), "v"(gaddr)
               : "memory");
}

template <int PHASE>
__global__ __launch_bounds__(TPB)
void forgetmult_chunk_kernel(const float* __restrict__ f,
                             const float* __restrict__ x,
                             const float* __restrict__ carry,   // PHASE==3
                             float* __restrict__ Aout,          // PHASE==1
                             float* __restrict__ Bout,          // PHASE==1
                             float* __restrict__ out) {         // PHASE==3
  // [buf][f/x][row][channel] : 2*2*4*512*4 = 32 KB
  __shared__ float smem[2][2][T_TILE][CH_PER_BLK];

  const int tid   = threadIdx.x;
  const int chunk = blockIdx.y;
  const int cpair = blockIdx.x * TPB + tid;   // float2 channel index
  const int ch0   = 2 * cpair;
  const int t0    = chunk * CHUNK_LEN;

  const float* fbase = f + (size_t)t0 * NCH + ch0;
  const float* xbase = x + (size_t)t0 * NCH + ch0;

  // Low 32 bits of a generic pointer to LDS == LDS byte offset (aperture in
  // the high half), which is exactly what the async op's dsaddr VGPR wants.
  const uint32_t lds_base = (uint32_t)(uintptr_t)(void*)&smem[0][0][0][0];
  const uint32_t lane_off = (uint32_t)(2u * (uint32_t)tid * sizeof(float));

  auto prefetch_tile = [&](int tile, int buf) {
    const float* fp = fbase + (size_t)tile * T_TILE * NCH;
    const float* xp = xbase + (size_t)tile * T_TILE * NCH;
#pragma unroll
    for (int r = 0; r < T_TILE; ++r) {
      const uint32_t lf = lds_base +
          (uint32_t)(((buf * 2 + 0) * T_TILE + r) * (CH_PER_BLK * 4)) + lane_off;
      const uint32_t lx = lds_base +
          (uint32_t)(((buf * 2 + 1) * T_TILE + r) * (CH_PER_BLK * 4)) + lane_off;
      async_copy_b64(lf, fp + (size_t)r * NCH);
      async_copy_b64(lx, xp + (size_t)r * NCH);
    }
  };

  float2 h  = make_float2(0.f, 0.f);
  float2 aa = make_float2(1.f, 1.f);
  float2 bb = make_float2(0.f, 0.f);
  if (PHASE == 3) {
    h = ((const float2*)(carry + (size_t)chunk * NCH))[cpair];
  }

  prefetch_tile(0, 0);

  for (int tile = 0; tile < N_TILES; ++tile) {
    const int buf = tile & 1;
    if (tile + 1 < N_TILES) {
      // make sure our DS reads of the buffer we are about to overwrite are done
      asm volatile("s_wait_dscnt 0" ::: "memory");
      prefetch_tile(tile + 1, buf ^ 1);
      // 8 newest DMAs (next tile) may stay in flight; this tile's 8 are done
      asm volatile("s_wait_asynccnt 8" ::: "memory");
    } else {
      asm volatile("s_wait_asynccnt 0" ::: "memory");
    }
#pragma unroll
    for (int r = 0; r < T_TILE; ++r) {
      // each wave reads only the LDS columns it DMA'd itself -> no barrier
      const float2 fv = ((const float2*)&smem[buf][0][r][0])[tid];
      const float2 xv = ((const float2*)&smem[buf][1][r][0])[tid];
      if (PHASE == 1) {
        bb.x = fv.x * xv.x + (1.f - fv.x) * bb.x;
        bb.y = fv.y * xv.y + (1.f - fv.y) * bb.y;
        aa.x *= (1.f - fv.x);
        aa.y *= (1.f - fv.y);
      } else {
        h.x = fv.x * xv.x + (1.f - fv.x) * h.x;
        h.y = fv.y * xv.y + (1.f - fv.y) * h.y;
        const int t = t0 + tile * T_TILE + r;
        ((float2*)(out + (size_t)t * NCH))[cpair] = h;
      }
    }
  }

  if (PHASE == 1) {
    ((float2*)(Aout + (size_t)chunk * NCH))[cpair] = aa;
    ((float2*)(Bout + (size_t)chunk * NCH))[cpair] = bb;
  }
}

// Tiny inter-chunk scan: h_in(chunk c+1) = A_c * h_in(c) + B_c per channel.
__global__ __launch_bounds__(256)
void carry_scan_kernel(const float* __restrict__ A,
                       const float* __restrict__ B,
                       const float* __restrict__ h0,
                       float* __restrict__ carry) {
  const int ch = blockIdx.x * 256 + threadIdx.x;
  float h = h0[ch];
  carry[ch] = h;
#pragma unroll
  for (int c = 0; c < CHUNKS - 1; ++c) {
    h = A[(size_t)c * NCH + ch] * h + B[(size_t)c * NCH + ch];
    carry[(size_t)(c + 1) * NCH + ch] = h;
  }
}

extern "C" void kernel_launch(void* const* d_in, const int* in_sizes, int n_in,
                              void* d_out, int out_size, void* d_ws, size_t ws_size,
                              hipStream_t stream) {
  (void)in_sizes; (void)n_in; (void)out_size; (void)ws_size;
  const float* f  = (const float*)d_in[0];          // (SEQ, BATCH, HIDDEN)
  const float* x  = (const float*)d_in[1];          // (SEQ, BATCH, HIDDEN)
  const float* h0 = (const float*)d_in[2];          // (BATCH, HIDDEN)
  float* out = (float*)d_out;                        // (SEQ, BATCH, HIDDEN)

  float* A     = (float*)d_ws;                       // CHUNKS*NCH floats
  float* B     = A + (size_t)CHUNKS * NCH;           // CHUNKS*NCH floats
  float* carry = B + (size_t)CHUNKS * NCH;           // CHUNKS*NCH floats
  // total workspace: 3 * 16 * 16384 * 4 B = 3 MiB

  dim3 grid(NCH / CH_PER_BLK, CHUNKS);               // (32, 16)

  forgetmult_chunk_kernel<1><<<grid, TPB, 0, stream>>>(f, x, nullptr, A, B, nullptr);
  carry_scan_kernel<<<NCH / 256, 256, 0, stream>>>(A, B, h0, carry);
  forgetmult_chunk_kernel<3><<<grid, TPB, 0, stream>>>(f, x, carry, nullptr, nullptr, out);
}